// LengthRegulator_3358664426438
// MI455X (gfx1250) — compile-verified
//
#include <hip/hip_runtime.h>
#include <hip/hip_bf16.h>
#include <stdint.h>

// ---------------------------------------------------------------------------
// Kernel 1: per-batch inclusive cumsum of durations (N = 1024).
// One wave32 per batch: each lane owns 32 contiguous elements in registers,
// serial prefix in-lane, then a 5-step shfl_up wave scan of lane totals.
// No LDS, no barriers.
// ---------------------------------------------------------------------------
__global__ __launch_bounds__(32) void lr_cumsum_kernel(
    const int* __restrict__ dur, int* __restrict__ cum, int N)
{
    const int b    = blockIdx.x;
    const int lane = threadIdx.x;

    const int4* src = (const int4*)(dur + (size_t)b * N + lane * 32);
    int v[32];

#pragma unroll
    for (int i = 0; i < 8; ++i) {
        int4 t = src[i];
        v[4 * i + 0] = t.x; v[4 * i + 1] = t.y;
        v[4 * i + 2] = t.z; v[4 * i + 3] = t.w;
    }

    // in-lane inclusive prefix over the 32 owned elements
    int sum = 0;
#pragma unroll
    for (int i = 0; i < 32; ++i) { sum += v[i]; v[i] = sum; }

    // wave-inclusive scan of lane totals -> exclusive offset for this lane
    int s = sum;
#pragma unroll
    for (int off = 1; off < 32; off <<= 1) {
        int n = __shfl_up(s, off, 32);
        if (lane >= off) s += n;
    }
    const int excl = s - sum;

    int4* dst = (int4*)(cum + (size_t)b * N + lane * 32);
#pragma unroll
    for (int i = 0; i < 8; ++i) {
        int4 o;
        o.x = v[4 * i + 0] + excl; o.y = v[4 * i + 1] + excl;
        o.z = v[4 * i + 2] + excl; o.w = v[4 * i + 3] + excl;
        dst[i] = o;
    }
}

// ---------------------------------------------------------------------------
// Kernel 2: expand rows. Tile = 32 output frames per block, 256 threads.
// Row copy goes memory -> LDS -> memory via CDNA5 async DMA (ASYNCcnt),
// never staging payload data through VGPRs.
//   loads : th:TH_LOAD_HT  (x has ~3.5x reuse and fits in 192MB L2 - protect it)
//   stores: th:TH_STORE_NT (201MB write-once stream - don't evict x from L2)
// ---------------------------------------------------------------------------
#define TTILE 32               // output frames per block
#define CHUNKS_PER_ROW 96      // 384 floats / 4 per b128 chunk
#define ELEMS (TTILE * CHUNKS_PER_ROW)   // 3072 chunks per tile
#define THREADS 256
#define ELEMS_PER_THREAD (ELEMS / THREADS) // 12

__global__ __launch_bounds__(THREADS) void lr_expand_kernel(
    const float* __restrict__ x,     // [B, N, 384]
    const int*   __restrict__ cum,   // [B, N] inclusive cumsum
    float*       __restrict__ out,   // [B, max_len, 384]
    float*       __restrict__ mask,  // [B, max_len] (0.0 / 1.0)
    int N, int max_len)
{
    __shared__ float4 buf[ELEMS];        // 48 KB staging tile (16B aligned)
    __shared__ int    scum[1024];        // 4 KB cumsum for this batch
    __shared__ int    srow[TTILE];       // clamped source row per frame
    __shared__ int    smask[TTILE];      // masked flag per frame

    const int b   = blockIdx.x;
    const int t0  = blockIdx.y * TTILE;
    const int tid = threadIdx.x;

    // --- stage cumsum into LDS (256 threads x int4 = 1024 ints) ---
    const int4* cum4 = (const int4*)(cum + (size_t)b * N);
    ((int4*)scum)[tid] = cum4[tid];
    __syncthreads();

    // --- binary search: searchsorted(cum, t, side='right') == upper_bound ---
    if (tid < TTILE) {
        const int t  = t0 + tid;
        int lo = 0, hi = N;
        while (lo < hi) {
            int mid = (lo + hi) >> 1;
            if (scum[mid] <= t) lo = mid + 1; else hi = mid;
        }
        const int m = (t >= scum[N - 1]) ? 1 : 0;
        srow[tid]  = (lo < N - 1) ? lo : (N - 1);
        smask[tid] = m;
        mask[(size_t)b * max_len + t] = m ? 1.0f : 0.0f;
    }
    __syncthreads();

    const unsigned lds_base = (unsigned)(size_t)(void*)buf; // low 32 bits = LDS offset
    const float* xb = x + (size_t)b * N * 384;
    float*       ob = out + ((size_t)b * max_len + t0) * 384;

    // --- async DMA: global -> LDS, one b128 chunk per (thread, k) ---
    // e = tid + k*256: consecutive lanes of a wave read contiguous 16B chunks.
#pragma unroll
    for (int k = 0; k < ELEMS_PER_THREAD; ++k) {
        const int e = tid + k * THREADS;
        const int r = e / CHUNKS_PER_ROW;
        const int l = e % CHUNKS_PER_ROW;
        const uint64_t src = (uint64_t)(size_t)(xb + (size_t)srow[r] * 384 + l * 4);
        const unsigned ldsa = lds_base + (unsigned)e * 16u;
        asm volatile("global_load_async_to_lds_b128 %0, %1, off th:TH_LOAD_HT"
                     :: "v"(ldsa), "v"(src) : "memory");
    }

    // Each wave's subsequent LDS touches are confined to bytes its own lanes
    // loaded above, so a per-wave ASYNCcnt wait suffices (no barrier).
    asm volatile("s_wait_asynccnt 0" ::: "memory");

    // --- zero masked rows in LDS (normal ds_store path, DScnt) ---
#pragma unroll
    for (int k = 0; k < ELEMS_PER_THREAD; ++k) {
        const int e = tid + k * THREADS;
        const int r = e / CHUNKS_PER_ROW;
        if (smask[r]) {
            buf[e] = make_float4(0.0f, 0.0f, 0.0f, 0.0f);
        }
    }
    // The async stores below read LDS; compiler can't see that dependency,
    // so drain DScnt explicitly before they issue.
    asm volatile("s_wait_dscnt 0" ::: "memory");

    // --- async DMA: LDS -> global (non-temporal streaming stores) ---
#pragma unroll
    for (int k = 0; k < ELEMS_PER_THREAD; ++k) {
        const int e = tid + k * THREADS;
        const int r = e / CHUNKS_PER_ROW;
        const int l = e % CHUNKS_PER_ROW;
        const uint64_t dst = (uint64_t)(size_t)(ob + (size_t)r * 384 + l * 4);
        const unsigned ldsa = lds_base + (unsigned)e * 16u;
        asm volatile("global_store_async_from_lds_b128 %0, %1, off th:TH_STORE_NT"
                     :: "v"(dst), "v"(ldsa) : "memory");
    }
    // S_ENDPGM performs an implicit wait-idle, but drain explicitly anyway.
    asm volatile("s_wait_asynccnt 0" ::: "memory");
}

// ---------------------------------------------------------------------------
// Host-side launcher.
// Inputs: d_in[0] = x (f32, B*N*D), d_in[1] = durations (i32, B*N),
//         d_in[2] = max_len scalar (on device; not readable during capture).
// Output: d_out = [out (B*max_len*D f32)] ++ [mel_mask (B*max_len f32)].
// ---------------------------------------------------------------------------
extern "C" void kernel_launch(void* const* d_in, const int* in_sizes, int n_in,
                              void* d_out, int out_size, void* d_ws, size_t ws_size,
                              hipStream_t stream)
{
    const float* x   = (const float*)d_in[0];
    const int*   dur = (const int*)d_in[1];

    const int N = 1024;                         // reference: N = 1024
    const int D = in_sizes[0] / in_sizes[1];    // 384
    const int B = in_sizes[1] / N;              // 32
    const int max_len = out_size / (B * (D + 1)); // 4096

    int*   cum  = (int*)d_ws;                            // B*N ints of scratch
    float* out  = (float*)d_out;                         // B*max_len*D
    float* mask = out + (size_t)B * max_len * D;         // B*max_len

    lr_cumsum_kernel<<<B, 32, 0, stream>>>(dur, cum, N);

    dim3 grid(B, max_len / TTILE);
    lr_expand_kernel<<<grid, THREADS, 0, stream>>>(x, cum, out, mask, N, max_len);
}